// Predict_84937273246072
// MI455X (gfx1250) — compile-verified
//
#include <hip/hip_runtime.h>
#include <math.h>

typedef __attribute__((ext_vector_type(2))) float v2f;
typedef __attribute__((ext_vector_type(4))) float v4f;
typedef __attribute__((ext_vector_type(8))) float v8f;
typedef __attribute__((ext_vector_type(4))) int   v4i;

// address-space-qualified element types for the async global->LDS builtin
typedef __attribute__((address_space(1))) v4i global_v4i;
typedef __attribute__((address_space(3))) v4i shared_v4i;

#define LOD 32
#define LSD 64
#define OUTW 160

// workspace layout (float offsets)
#define OFF_BW1 0          // 16 k4 * 4 n * 32 lanes * 2 = 4096 floats
#define OFF_BW2 4096       // 24 k4 * 6 n * 32 lanes * 2 = 9216 floats
#define OFF_BTM 13312      // 64
#define OFF_BSQ 13376      // 64
#define OFF_Q   13440      // 64
#define WS_FLOATS 13504    // = 3376 float4 (54,016 bytes)

#if __has_builtin(__builtin_amdgcn_global_load_async_to_lds_b128) && \
    __has_builtin(__builtin_amdgcn_s_wait_asynccnt)
#define USE_ASYNC_LDS 1
#else
#define USE_ASYNC_LDS 0
#endif

// ---------------------------------------------------------------------------
// Prep kernel: build WMMA-B-fragment-swizzled weight matrices + constants.
// B fragment layout (f32 4x16): VGPR v, lane L -> K = 4*k4 + v + 2*(L/16),
//                                                N = 16*n + (L%16)
// ---------------------------------------------------------------------------
__global__ void prep_kernel(const float* __restrict__ tm11, const float* __restrict__ tm12,
                            const float* __restrict__ tm21, const float* __restrict__ tm22,
                            const float* __restrict__ b_tm, const float* __restrict__ log_pn,
                            float* __restrict__ ws) {
  int idx = blockIdx.x * blockDim.x + threadIdx.x;
  if (idx >= WS_FLOATS) return;
  float val = 0.0f;
  if (idx < OFF_BW2) {
    // GEMM1 weights: W1T[k][j], X1 = [mu|ml], out cols j: [nmu|nml]
    int t = idx;
    int v = t & 1;  t >>= 1;
    int L = t & 31; t >>= 5;
    int n = t & 3;  int k4 = t >> 2;
    int k = k4 * 4 + v + 2 * (L >> 4);
    int j = n * 16 + (L & 15);
    if (j < LOD) {
      val = (k < LOD) ? tm11[j * LOD + k] : tm12[j * LOD + (k - LOD)];
    } else {
      int jj = j - LOD;
      val = (k < LOD) ? tm21[jj * LOD + k] : tm22[jj * LOD + (k - LOD)];
    }
  } else if (idx < OFF_BTM) {
    // GEMM2 weights: W2T[k][j], X2 = [cu0|cs0|cl0], out cols j: [ncu|ncl|ncs]
    int t = idx - OFF_BW2;
    int v = t & 1;  t >>= 1;
    int L = t & 31; t >>= 5;
    int n = t % 6;  int k4 = t / 6;
    int k = k4 * 4 + v + 2 * (L >> 4);
    int j = n * 16 + (L & 15);
    int seg = k >> 5;          // 0: cu term, 1: cs term, 2: cl term
    int kk = k & (LOD - 1);
    if (j < LOD) {                       // ncu row
      int o = j * LOD + kk;
      float a = tm11[o], b = tm12[o];
      val = (seg == 0) ? a * a : (seg == 1) ? 2.0f * a * b : b * b;
    } else if (j < 2 * LOD) {            // ncl row
      int o = (j - LOD) * LOD + kk;
      float a = tm21[o], b = tm22[o];
      val = (seg == 0) ? a * a : (seg == 1) ? 2.0f * a * b : b * b;
    } else {                             // ncs row
      int o = (j - 2 * LOD) * LOD + kk;
      float t11 = tm11[o], t12 = tm12[o], t21 = tm21[o], t22 = tm22[o];
      val = (seg == 0) ? t21 * t11
          : (seg == 1) ? (t22 * t11 + t21 * t12)
                       : t22 * t12;
    }
  } else if (idx < OFF_BSQ) {
    val = b_tm[idx - OFF_BTM];
  } else if (idx < OFF_Q) {
    float c = b_tm[idx - OFF_BSQ];
    val = c * c;
  } else {
    float x = log_pn[idx - OFF_Q];
    val = (x < 0.0f) ? __expf(x) : (x + 1.0f);   // elup1
  }
  ws[idx] = val;
}

// ---------------------------------------------------------------------------
// Main kernel: one wave32 per 16 rows. Weights staged to LDS once per block
// (async global->LDS copy when available), then two register-resident WMMA
// GEMMs with B fragments from LDS (conflict-free ds_load_b64).
// ---------------------------------------------------------------------------
__global__ void __launch_bounds__(256)
predict_kernel(const float* __restrict__ pm0, const float* __restrict__ cu0,
               const float* __restrict__ cl0, const float* __restrict__ cs0,
               const float* __restrict__ pm1, const float* __restrict__ cov1,
               const float* __restrict__ ws, float* __restrict__ out, int nrows) {
  __shared__ __align__(16) float smem[WS_FLOATS];

  // ---- stage swizzled weights + constants into LDS (once per block) ----
  {
    v4i* gsrc = (v4i*)ws;       // const dropped: builtin takes non-const ptr
    v4i* ldst = (v4i*)smem;
#if USE_ASYNC_LDS
    for (int i = threadIdx.x; i < WS_FLOATS / 4; i += 256) {
      __builtin_amdgcn_global_load_async_to_lds_b128(
          (global_v4i*)(gsrc + i), (shared_v4i*)(ldst + i), 0, 0);
    }
    __builtin_amdgcn_s_wait_asynccnt(0);
#else
    for (int i = threadIdx.x; i < WS_FLOATS / 4; i += 256) ldst[i] = gsrc[i];
#endif
    __syncthreads();
  }

  const int lane = threadIdx.x & 31;
  const int wave = threadIdx.x >> 5;
  const int r0 = (blockIdx.x * 8 + wave) * 16;
  if (r0 >= nrows) return;                  // wave-uniform: EXEC stays all-ones
  const int hi = lane >> 4;                 // 0: lanes 0-15, 1: lanes 16-31
  const int lo = lane & 15;
  const int arow = r0 + lo;                 // A-fragment row for this lane

  // ================= GEMM1: mean(16x64) = X1(16x64) @ W1T(64x64) ============
  v8f acc[4];
#pragma unroll
  for (int n = 0; n < 4; ++n) { v8f z = {}; acc[n] = z; }
  {
    const float* abase = pm0 + (size_t)arow * LSD + hi * 2;
    const float* bbase = smem + OFF_BW1 + lane * 2;
#pragma unroll
    for (int k4 = 0; k4 < 16; ++k4) {
      v2f a = *(const v2f*)(abase + k4 * 4);   // lane holds K = 4k4+2hi, +1
#pragma unroll
      for (int n = 0; n < 4; ++n) {
        v2f b = *(const v2f*)(bbase + (k4 * 4 + n) * 64);
        acc[n] = __builtin_amdgcn_wmma_f32_16x16x4_f32(
            false, a, false, b, (short)0, acc[n], false, false);
      }
    }
  }
  // epilogue: mean += b_tm * post_mean1 ; store out cols [0,64)
#pragma unroll
  for (int n = 0; n < 4; ++n) {
    int col = n * 16 + lo;
    float btm = smem[OFF_BTM + col];
#pragma unroll
    for (int v = 0; v < 8; ++v) {
      int rw = r0 + v + 8 * hi;             // C/D layout: M = v + 8*hi
      out[(size_t)rw * OUTW + col] =
          acc[n][v] + btm * pm1[(size_t)rw * LSD + col];
    }
  }

  // ================= GEMM2: cov(16x96) = X2(16x96) @ W2T(96x96) =============
  v8f acc2[6];
#pragma unroll
  for (int n = 0; n < 6; ++n) { v8f z = {}; acc2[n] = z; }
  {
    const float* bbase = smem + OFF_BW2 + lane * 2;
#pragma unroll
    for (int k4 = 0; k4 < 24; ++k4) {
      const float* src = (k4 < 8) ? cu0 : (k4 < 16) ? cs0 : cl0;
      int lk = (k4 & 7) * 4 + hi * 2;
      v2f a = *(const v2f*)(src + (size_t)arow * LOD + lk);
#pragma unroll
      for (int n = 0; n < 6; ++n) {
        v2f b = *(const v2f*)(bbase + (k4 * 6 + n) * 64);
        acc2[n] = __builtin_amdgcn_wmma_f32_16x16x4_f32(
            false, a, false, b, (short)0, acc2[n], false, false);
      }
    }
  }
  // epilogue: ncu/ncl += b_tm^2*cov1 + elup1(log_pn); ncs passthrough.
  // out cols: ncu -> [64,96), ncl -> [96,128), ncs -> [128,160)
#pragma unroll
  for (int n = 0; n < 6; ++n) {
    int colj = n * 16 + lo;                 // 0..95 within [ncu|ncl|ncs]
    if (n < 4) {                            // ncu / ncl (colj < 64), uniform
      float bsq = smem[OFF_BSQ + colj];
      float q   = smem[OFF_Q + colj];
#pragma unroll
      for (int v = 0; v < 8; ++v) {
        int rw = r0 + v + 8 * hi;
        out[(size_t)rw * OUTW + LSD + colj] =
            acc2[n][v] + bsq * cov1[(size_t)rw * LSD + colj] + q;
      }
    } else {                                // ncs
#pragma unroll
      for (int v = 0; v < 8; ++v) {
        int rw = r0 + v + 8 * hi;
        out[(size_t)rw * OUTW + LSD + colj] = acc2[n][v];
      }
    }
  }
}

// ---------------------------------------------------------------------------
extern "C" void kernel_launch(void* const* d_in, const int* in_sizes, int n_in,
                              void* d_out, int out_size, void* d_ws, size_t ws_size,
                              hipStream_t stream) {
  const float* pm0    = (const float*)d_in[0];
  const float* cu0    = (const float*)d_in[1];
  const float* cl0    = (const float*)d_in[2];
  const float* cs0    = (const float*)d_in[3];
  const float* pm1    = (const float*)d_in[4];
  const float* cov1   = (const float*)d_in[5];
  const float* tm11   = (const float*)d_in[6];
  const float* tm12   = (const float*)d_in[7];
  const float* tm21   = (const float*)d_in[8];
  const float* tm22   = (const float*)d_in[9];
  const float* b_tm   = (const float*)d_in[10];
  const float* log_pn = (const float*)d_in[11];
  float* ws  = (float*)d_ws;
  float* out = (float*)d_out;

  const int nrows = in_sizes[0] / LSD;      // B = 262144

  prep_kernel<<<(WS_FLOATS + 255) / 256, 256, 0, stream>>>(
      tm11, tm12, tm21, tm22, b_tm, log_pn, ws);

  const int nblocks = (nrows + 127) / 128;  // 8 waves * 16 rows per block
  predict_kernel<<<nblocks, 256, 0, stream>>>(
      pm0, cu0, cl0, cs0, pm1, cov1, ws, out, nrows);
}